// SoftCrossEntropyLoss_60559038873702
// MI455X (gfx1250) — compile-verified
//
#include <hip/hip_runtime.h>
#include <hip/hip_bf16.h>
#include <stdint.h>

// Soft cross-entropy loss, HBM-bandwidth-bound streaming reduction.
// CDNA5 path: double-buffered GLOBAL_LOAD_ASYNC_TO_LDS_B128 staging + s_wait_asynccnt.

#define TPB    256
#define TILE   2048            // floats per tile per array (8 KB); 2 arrays x 2 buffers = 32 KB LDS
#define NWAVES (TPB / 32)

__device__ __forceinline__ void async_load_b128(uint32_t lds_addr, uint32_t voff, const void* sbase) {
  // GVS mode: mem_addr = SADDR(u64) + VADDR(i32); VDST = LDS byte address.
  asm volatile("global_load_async_to_lds_b128 %0, %1, %2"
               :: "v"(lds_addr), "v"(voff), "s"(sbase)
               : "memory");
}

__device__ __forceinline__ void wait_async0() {
  asm volatile("s_wait_asynccnt 0" ::: "memory");
}

__global__ __launch_bounds__(TPB)
void soft_ce_row_kernel(const float* __restrict__ x,
                        const float* __restrict__ tg,
                        const float* __restrict__ w,
                        float* __restrict__ row_loss,
                        int K) {
  __shared__ __align__(16) float sx[2][TILE];
  __shared__ __align__(16) float st[2][TILE];
  __shared__ float red_m[NWAVES], red_s[NWAVES], red_a[NWAVES], red_b[NWAVES];

  const int row = blockIdx.x;
  const int tid = threadIdx.x;
  const float* xrow = x  + (size_t)row * (size_t)K;
  const float* trow = tg + (size_t)row * (size_t)K;

  // Low 32 bits of a flat pointer to __shared__ == wave-relative LDS byte address.
  const uint32_t lds_x0 = (uint32_t)(uintptr_t)&sx[0][0];
  const uint32_t lds_x1 = (uint32_t)(uintptr_t)&sx[1][0];
  const uint32_t lds_t0 = (uint32_t)(uintptr_t)&st[0][0];
  const uint32_t lds_t1 = (uint32_t)(uintptr_t)&st[1][0];

  const int ntiles = (K + TILE - 1) / TILE;

  // Prologue: stage tile 0 into buffer 0.
  {
    const int cnt  = (K < TILE) ? K : TILE;
    const int cnt4 = cnt >> 2;
    for (int j = tid; j < cnt4; j += TPB) {
      const uint32_t go = (uint32_t)(j * 16);
      async_load_b128(lds_x0 + (uint32_t)(j * 16), go, xrow);
      async_load_b128(lds_t0 + (uint32_t)(j * 16), go, trow);
    }
  }
  wait_async0();
  __syncthreads();

  float m = -3.402823466e+38f;   // running lane-local max
  float s = 0.0f;                // running lane-local sum of exp(x - m)
  float A = 0.0f;                // sum t*w*x
  float B = 0.0f;                // sum t*w

  for (int i = 0; i < ntiles; ++i) {
    const int buf  = i & 1;
    const int base = i * TILE;
    const int cnt  = (K - base < TILE) ? (K - base) : TILE;
    const int cnt4 = cnt >> 2;

    // Prefetch tile i+1 into the other buffer (overlaps with compute below).
    if (i + 1 < ntiles) {
      const int nbase = base + TILE;
      const int ncnt  = (K - nbase < TILE) ? (K - nbase) : TILE;
      const int ncnt4 = ncnt >> 2;
      const uint32_t lx = buf ? lds_x0 : lds_x1;
      const uint32_t lt = buf ? lds_t0 : lds_t1;
      for (int j = tid; j < ncnt4; j += TPB) {
        const uint32_t go = (uint32_t)((nbase >> 2) + j) * 16u;
        async_load_b128(lx + (uint32_t)(j * 16), go, xrow);
        async_load_b128(lt + (uint32_t)(j * 16), go, trow);
      }
    }

    const float4* sx4 = (const float4*)&sx[buf][0];
    const float4* st4 = (const float4*)&st[buf][0];
    const float4* w4  = (const float4*)(w + base);

    // Pass 1 over LDS: lane-local tile max (cheap), one rescale exp per tile.
    float tmax = -3.402823466e+38f;
    for (int j = tid; j < cnt4; j += TPB) {
      const float4 v = sx4[j];
      tmax = fmaxf(tmax, fmaxf(fmaxf(v.x, v.y), fmaxf(v.z, v.w)));
    }
    const float mn = fmaxf(m, tmax);
    s *= __expf(m - mn);         // finite init => underflows to 0 when s==0
    m  = mn;

    // Pass 2 over LDS: exp-sum + weighted accumulations (1 exp per element).
    for (int j = tid; j < cnt4; j += TPB) {
      const float4 xv = sx4[j];
      const float4 tv = st4[j];
      const float4 wv = w4[j];
      s += __expf(xv.x - m) + __expf(xv.y - m) + __expf(xv.z - m) + __expf(xv.w - m);
      const float tw0 = tv.x * wv.x, tw1 = tv.y * wv.y;
      const float tw2 = tv.z * wv.z, tw3 = tv.w * wv.w;
      B += tw0 + tw1 + tw2 + tw3;
      A = fmaf(tw0, xv.x, A); A = fmaf(tw1, xv.y, A);
      A = fmaf(tw2, xv.z, A); A = fmaf(tw3, xv.w, A);
    }

    // Wait for the prefetched tile (only outstanding async ops), then sync waves.
    wait_async0();
    __syncthreads();
  }

  // Wave32 reduction of (m, s, A, B).
  for (int off = 16; off > 0; off >>= 1) {
    const float m2 = __shfl_xor(m, off, 32);
    const float s2 = __shfl_xor(s, off, 32);
    const float A2 = __shfl_xor(A, off, 32);
    const float B2 = __shfl_xor(B, off, 32);
    const float mm = fmaxf(m, m2);
    s = s * __expf(m - mm) + s2 * __expf(m2 - mm);
    m = mm;
    A += A2;
    B += B2;
  }
  const int wave = tid >> 5;
  if ((tid & 31) == 0) {
    red_m[wave] = m; red_s[wave] = s; red_a[wave] = A; red_b[wave] = B;
  }
  __syncthreads();
  if (tid == 0) {
    float fm = red_m[0], fs = red_s[0], fa = red_a[0], fb = red_b[0];
    for (int wv = 1; wv < NWAVES; ++wv) {
      const float mm = fmaxf(fm, red_m[wv]);
      fs = fs * __expf(fm - mm) + red_s[wv] * __expf(red_m[wv] - mm);
      fm = mm;
      fa += red_a[wv];
      fb += red_b[wv];
    }
    // loss_n = (max + log(sum exp)) * B - A
    row_loss[row] = (fm + logf(fs)) * fb - fa;
  }
}

__global__ __launch_bounds__(TPB)
void soft_ce_final_reduce(const float* __restrict__ row_loss,
                          float* __restrict__ out,
                          int n, float inv_n) {
  __shared__ float sh[TPB];
  float acc = 0.0f;
  for (int i = threadIdx.x; i < n; i += TPB) acc += row_loss[i];
  sh[threadIdx.x] = acc;
  __syncthreads();
  for (int off = TPB / 2; off > 0; off >>= 1) {
    if (threadIdx.x < off) sh[threadIdx.x] += sh[threadIdx.x + off];
    __syncthreads();
  }
  if (threadIdx.x == 0) out[0] = sh[0] * inv_n;
}

extern "C" void kernel_launch(void* const* d_in, const int* in_sizes, int n_in,
                              void* d_out, int out_size, void* d_ws, size_t ws_size,
                              hipStream_t stream) {
  const float* x  = (const float*)d_in[0];   // input  [N, K] fp32
  const float* tg = (const float*)d_in[1];   // target [N, K] fp32
  const float* w  = (const float*)d_in[2];   // class_weights [K] fp32
  float* out = (float*)d_out;                // scalar fp32

  const int K = in_sizes[2];
  const int N = in_sizes[0] / K;

  float* row_loss = (float*)d_ws;            // N floats of scratch (32 KB)

  soft_ce_row_kernel<<<N, TPB, 0, stream>>>(x, tg, w, row_loss, K);
  soft_ce_final_reduce<<<1, TPB, 0, stream>>>(row_loss, out, N, 1.0f / (float)N);
}